// Linear_36146444763339
// MI455X (gfx1250) — compile-verified
//
#include <hip/hip_runtime.h>

typedef __attribute__((ext_vector_type(2))) float v2f;
typedef __attribute__((ext_vector_type(4))) float v4f;
typedef __attribute__((ext_vector_type(8))) float v8f;

#define MUL   128
#define FEAT  1152
#define ALPHA 0.08838834764831845f   /* 1/sqrt(128) */

// Max LDS chunk: 16 rows x (128*5 + 4 pad) floats = 41216 bytes
#define MAX_P (128 * 5 + 4)

// Stage a full 16 x (128*D) f32 chunk of x (starting at column `off`) into LDS,
// padded row stride P = 128*D + 4 dwords (P % 64 == 4 -> conflict-free gathers).
template <int D>
__device__ inline void stage_tile(float* lds, const float* __restrict__ x,
                                  int n0, int off, int tid) {
    const int C4 = (128 * D) >> 2;     // float4 per row
    const int P  = 128 * D + 4;
#pragma unroll 2
    for (int idx = tid; idx < 16 * C4; idx += 256) {
        const int row = idx / C4;
        const int c4  = idx - row * C4;
        v4f v = __builtin_nontemporal_load(
            (const v4f*)(x + (n0 + row) * FEAT + off + c4 * 4));
        *(v4f*)(lds + row * P + c4 * 4) = v;
    }
}

// One weight block: this wave computes its 16-wide w-tile for all D components.
// A: 16x4 f32 frag  (VGPR0: K=khalf, VGPR1: K=khalf+1; lanes 0-15 M, 16-31 K+2)
// B: 4x16 f32 frag  (mirror layout over N columns)
// C/D: 8 VGPRs, lanes 0-15 -> M=r, lanes 16-31 -> M=r+8, N = lane%16
template <int D>
__device__ inline void compute_block(const float* lds, const float* __restrict__ w,
                                     float* __restrict__ out, int n0,
                                     int off, int lane, int wv) {
    const int P     = 128 * D + 4;
    const int m     = lane & 15;           // A-row / B,C-column index
    const int khalf = (lane >> 4) << 1;    // 0 for lanes 0-15, 2 for 16-31
    const int wbase = wv * 16;

    v8f acc[D] = {};

#pragma unroll 4
    for (int s = 0; s < 32; ++s) {
        const int k0 = 4 * s + khalf;
        v2f b;
        b.x = w[(k0 + 0) * MUL + wbase + m];
        b.y = w[(k0 + 1) * MUL + wbase + m];
#pragma unroll
        for (int i = 0; i < D; ++i) {
            v2f a;
            a.x = lds[m * P + (k0 + 0) * D + i];
            a.y = lds[m * P + (k0 + 1) * D + i];
            acc[i] = __builtin_amdgcn_wmma_f32_16x16x4_f32(
                false, a, false, b, (short)0, acc[i], false, false);
        }
    }

    // Store: one base address per component, 8 unguarded stores with immediate
    // row offsets (r*FEAT*4 fits in the 24-bit IOFFSET). Streaming (NT) hint.
    const int mbase   = (lane < 16) ? 0 : 8;     // first of this lane's 8 rows
    const int colbase = off + (wbase + m) * D;
#pragma unroll
    for (int i = 0; i < D; ++i) {
        float* p = out + (n0 + mbase) * FEAT + colbase + i;
#pragma unroll
        for (int r = 0; r < 8; ++r)
            __builtin_nontemporal_store(acc[i][r] * ALPHA, p + r * FEAT);
    }
}

// Full-tile kernel: every block owns exactly 16 rows. No guards anywhere.
__global__ __launch_bounds__(256) void eqlin_wmma_kernel(
    const float* __restrict__ x,
    const float* __restrict__ w0,
    const float* __restrict__ w1,
    const float* __restrict__ w2,
    float* __restrict__ out) {
    __shared__ float lds[16 * MAX_P];

    const int tid  = threadIdx.x;
    const int lane = tid & 31;
    const int wv   = tid >> 5;           // wave id 0..7 == w-tile
    const int n0   = blockIdx.x * 16;

    // Irrep block 0: d=1, cols [0,128)
    stage_tile<1>(lds, x, n0, /*off=*/0, tid);
    __syncthreads();
    compute_block<1>(lds, w0, out, n0, /*off=*/0, lane, wv);
    __syncthreads();

    // Irrep block 1: d=3, cols [128,512)
    stage_tile<3>(lds, x, n0, /*off=*/128, tid);
    __syncthreads();
    compute_block<3>(lds, w1, out, n0, /*off=*/128, lane, wv);
    __syncthreads();

    // Irrep block 2: d=5, cols [512,1152)
    stage_tile<5>(lds, x, n0, /*off=*/512, tid);
    __syncthreads();
    compute_block<5>(lds, w2, out, n0, /*off=*/512, lane, wv);
}

// Scalar tail kernel for n % 16 leftover rows (never launched when n % 16 == 0).
__global__ __launch_bounds__(256) void eqlin_tail_kernel(
    const float* __restrict__ x,
    const float* __restrict__ w0,
    const float* __restrict__ w1,
    const float* __restrict__ w2,
    float* __restrict__ out, int row0, int n) {
    const int t   = blockIdx.x * 256 + threadIdx.x;
    const int row = row0 + t / FEAT;
    const int col = t % FEAT;
    if (row >= n) return;

    const float* w;
    int d, off, wc, i;
    if (col < 128) {
        w = w0; d = 1; off = 0;   wc = col;  i = 0;
    } else if (col < 512) {
        w = w1; d = 3; off = 128; const int c = col - 128; wc = c / 3; i = c - wc * 3;
    } else {
        w = w2; d = 5; off = 512; const int c = col - 512; wc = c / 5; i = c - wc * 5;
    }

    const float* xr = x + row * FEAT + off + i;
    float acc = 0.f;
    for (int u = 0; u < MUL; ++u)
        acc += xr[u * d] * w[u * MUL + wc];
    out[row * FEAT + col] = acc * ALPHA;
}

extern "C" void kernel_launch(void* const* d_in, const int* in_sizes, int n_in,
                              void* d_out, int out_size, void* d_ws, size_t ws_size,
                              hipStream_t stream) {
    const float* x  = (const float*)d_in[0];
    const float* w0 = (const float*)d_in[1];
    const float* w1 = (const float*)d_in[2];
    const float* w2 = (const float*)d_in[3];
    float* out = (float*)d_out;

    const int n      = in_sizes[0] / FEAT;
    const int ntiles = n / 16;           // full 16-row tiles
    const int rem    = n - ntiles * 16;

    if (ntiles > 0)
        eqlin_wmma_kernel<<<ntiles, 256, 0, stream>>>(x, w0, w1, w2, out);
    if (rem > 0) {
        const int tthreads = rem * FEAT;
        eqlin_tail_kernel<<<(tthreads + 255) / 256, 256, 0, stream>>>(
            x, w0, w1, w2, out, ntiles * 16, n);
    }
}